// GNNResNetBlock_Base_3435973837210
// MI455X (gfx1250) — compile-verified
//
#include <hip/hip_runtime.h>
#include <cstdint>

#define N_NODES 50000
#define NEDGE   800000
#define D       128
#define LN_EPS  1e-5f
#define NEG_SLOPE 0.01f

typedef __attribute__((ext_vector_type(2))) float v2f;
typedef __attribute__((ext_vector_type(8))) float v8f;

// ---------------------------------------------------------------------------
// Kernel 1: LayerNorm, one wave (32 lanes) per node, 4 feats/lane.
// Also zeroes the scatter accumulator for this node and seeds deg = 1 (self-loop).
// ---------------------------------------------------------------------------
__global__ __launch_bounds__(256) void ln_kernel(const float* __restrict__ x,
                                                 const float* __restrict__ ln_w,
                                                 const float* __restrict__ ln_b,
                                                 float* __restrict__ xn,
                                                 float* __restrict__ deg,
                                                 float* __restrict__ accum) {
    const int wave = threadIdx.x >> 5;
    const int lane = threadIdx.x & 31;
    const int node = blockIdx.x * 8 + wave;
    if (node >= N_NODES) return;

    float4 v = reinterpret_cast<const float4*>(x + (size_t)node * D)[lane];
    float s  = v.x + v.y + v.z + v.w;
    float ss = v.x * v.x + v.y * v.y + v.z * v.z + v.w * v.w;
    #pragma unroll
    for (int off = 16; off > 0; off >>= 1) {
        s  += __shfl_xor(s,  off, 32);
        ss += __shfl_xor(ss, off, 32);
    }
    const float mu  = s * (1.0f / D);
    float var       = ss * (1.0f / D) - mu * mu;
    var             = fmaxf(var, 0.0f);
    const float inv = rsqrtf(var + LN_EPS);

    float4 w4 = reinterpret_cast<const float4*>(ln_w)[lane];
    float4 b4 = reinterpret_cast<const float4*>(ln_b)[lane];
    float4 o;
    o.x = (v.x - mu) * inv * w4.x + b4.x;
    o.y = (v.y - mu) * inv * w4.y + b4.y;
    o.z = (v.z - mu) * inv * w4.z + b4.z;
    o.w = (v.w - mu) * inv * w4.w + b4.w;
    reinterpret_cast<float4*>(xn + (size_t)node * D)[lane] = o;

    float4 z = {0.f, 0.f, 0.f, 0.f};
    reinterpret_cast<float4*>(accum + (size_t)node * D)[lane] = z;
    if (lane == 0) deg[node] = 1.0f;   // self-loop weight
}

// ---------------------------------------------------------------------------
// Kernel 2: weighted in-degree: deg[col[e]] += ew[e]  (hardware f32 atomic)
// ---------------------------------------------------------------------------
__global__ __launch_bounds__(256) void deg_kernel(const long long* __restrict__ ei,
                                                  const float* __restrict__ ew,
                                                  float* __restrict__ deg) {
    int e = blockIdx.x * blockDim.x + threadIdx.x;
    if (e < NEDGE) {
        long long c = ei[(size_t)NEDGE + e];   // edgeIndex[1][e] (target)
        unsafeAtomicAdd(deg + c, ew[e]);       // -> global_atomic_add_f32
    }
}

// ---------------------------------------------------------------------------
// Kernel 3: deg -> dinv (in place)
// ---------------------------------------------------------------------------
__global__ __launch_bounds__(256) void dinv_kernel(float* __restrict__ deg) {
    int i = blockIdx.x * blockDim.x + threadIdx.x;
    if (i < N_NODES) {
        float d = deg[i];
        deg[i] = (d > 0.0f) ? rsqrtf(d) : 0.0f;
    }
}

// ---------------------------------------------------------------------------
// Kernel 4: fused dual GEMM  h = xn @ W_gcn, t = xn @ W_lin  (t -> d_out)
// 256 threads = 8 waves. Block owns a 16-node M tile staged in LDS (stride-132
// padded to dodge bank conflicts). Wave w owns N columns [16w,16w+16).
// K loop in steps of 4 with V_WMMA_F32_16X16X4_F32; A frag shared by both GEMMs.
// ---------------------------------------------------------------------------
__global__ __launch_bounds__(256) void gemm_kernel(const float* __restrict__ xn,
                                                   const float* __restrict__ Wg,
                                                   const float* __restrict__ Wl,
                                                   float* __restrict__ h,
                                                   float* __restrict__ tout) {
    __shared__ float tile[16 * 132];
    const int tileBase = blockIdx.x * 16;
    const int t = threadIdx.x;

    // Cooperative load of the 16x128 A tile: 8 floats per thread.
    {
        int elem = t * 8;
        int r = elem >> 7;       // /128
        int c = elem & 127;
        const float4* src = reinterpret_cast<const float4*>(xn + (size_t)(tileBase + r) * D + c);
        float4 a0 = src[0];
        float4 a1 = src[1];
        float* dst = &tile[r * 132 + c];
        reinterpret_cast<float4*>(dst)[0] = a0;
        reinterpret_cast<float4*>(dst)[1] = a1;
    }
    __syncthreads();

    const int wave    = t >> 5;
    const int lane    = t & 31;
    const int colBase = wave * 16;
    const int mrow    = lane & 15;          // A row for lanes 0-15 / 16-31
    const int ksel    = (lane >> 4) << 1;   // 0 for lanes 0-15, 2 for lanes 16-31
    const int col     = colBase + mrow;     // B column (lane&15)
    const float* trow = &tile[mrow * 132];

    v8f accg = {};
    v8f accl = {};
    #pragma unroll 4
    for (int k = 0; k < D; k += 4) {
        const int ka = k + ksel;
        v2f a;
        a.x = trow[ka];
        a.y = trow[ka + 1];
        v2f bg, bl;
        bg.x = Wg[(size_t)ka * D + col];
        bg.y = Wg[(size_t)(ka + 1) * D + col];
        bl.x = Wl[(size_t)ka * D + col];
        bl.y = Wl[(size_t)(ka + 1) * D + col];
        accg = __builtin_amdgcn_wmma_f32_16x16x4_f32(false, a, false, bg, (short)0, accg, false, false);
        accl = __builtin_amdgcn_wmma_f32_16x16x4_f32(false, a, false, bl, (short)0, accl, false, false);
    }

    // D layout: VGPR v, lanes 0-15 -> M=v,  N=lane;  lanes 16-31 -> M=8+v, N=lane-16
    const int mBase = (lane >> 4) << 3;
    const int n     = lane & 15;
    #pragma unroll
    for (int v = 0; v < 8; ++v) {
        const size_t idx = (size_t)(tileBase + mBase + v) * D + colBase + n;
        h[idx]    = accg[v];
        tout[idx] = accl[v];
    }
}

// ---------------------------------------------------------------------------
// Kernel 5: edge scatter. One wave per edge, 4 feats/lane (float4 gather,
// 4 hardware f32 atomic adds). accum[col] += h[row] * (dinv[row]*ew*dinv[col])
// ---------------------------------------------------------------------------
__global__ __launch_bounds__(256) void scatter_kernel(const long long* __restrict__ ei,
                                                      const float* __restrict__ ew,
                                                      const float* __restrict__ dinv,
                                                      const float* __restrict__ h,
                                                      float* __restrict__ accum) {
    const long long idx = (long long)blockIdx.x * blockDim.x + threadIdx.x;
    const int e    = (int)(idx >> 5);
    const int lane = (int)(idx & 31);
    if (e >= NEDGE) return;

    const long long r = ei[e];                    // source j
    const long long c = ei[(size_t)NEDGE + e];    // target i
    const float norm  = dinv[r] * ew[e] * dinv[c];

    const float4 hv = reinterpret_cast<const float4*>(h + (size_t)r * D)[lane];
    float* dst = accum + (size_t)c * D + lane * 4;
    unsafeAtomicAdd(dst + 0, hv.x * norm);   // -> global_atomic_add_f32 (no return)
    unsafeAtomicAdd(dst + 1, hv.y * norm);
    unsafeAtomicAdd(dst + 2, hv.z * norm);
    unsafeAtomicAdd(dst + 3, hv.w * norm);
}

// ---------------------------------------------------------------------------
// Kernel 6: epilogue. g = accum + h*dinv^2 (self-loop) + b_gcn; LeakyReLU;
// out = leaky(g) + t + b_lin   (t already stored in d_out by gemm_kernel)
// ---------------------------------------------------------------------------
__global__ __launch_bounds__(256) void final_kernel(const float* __restrict__ accum,
                                                    const float* __restrict__ h,
                                                    const float* __restrict__ dinv,
                                                    const float* __restrict__ b_gcn,
                                                    const float* __restrict__ b_lin,
                                                    float* __restrict__ out) {
    const int i = blockIdx.x * blockDim.x + threadIdx.x;
    if (i >= N_NODES * D) return;
    const int node = i >> 7;
    const int f    = i & 127;
    const float di = dinv[node];
    const float g  = accum[i] + h[i] * di * di + b_gcn[f];
    const float lr = (g >= 0.0f) ? g : NEG_SLOPE * g;
    out[i] = lr + out[i] + b_lin[f];
}

// ---------------------------------------------------------------------------
extern "C" void kernel_launch(void* const* d_in, const int* in_sizes, int n_in,
                              void* d_out, int out_size, void* d_ws, size_t ws_size,
                              hipStream_t stream) {
    const float*     x   = (const float*)d_in[0];
    const long long* ei  = (const long long*)d_in[1];   // int64 [2, E]
    const float*     ew  = (const float*)d_in[2];
    const float*     lnw = (const float*)d_in[3];
    const float*     lnb = (const float*)d_in[4];
    const float*     Wg  = (const float*)d_in[5];
    const float*     bg  = (const float*)d_in[6];
    const float*     Wl  = (const float*)d_in[7];
    const float*     bl  = (const float*)d_in[8];
    float* out = (float*)d_out;

    char* ws = (char*)d_ws;
    float* xn    = (float*)ws;  ws += (size_t)N_NODES * D * sizeof(float);
    float* h     = (float*)ws;  ws += (size_t)N_NODES * D * sizeof(float);
    float* accum = (float*)ws;  ws += (size_t)N_NODES * D * sizeof(float);
    float* deg   = (float*)ws;  ws += (size_t)N_NODES * sizeof(float);

    ln_kernel<<<(N_NODES + 7) / 8, 256, 0, stream>>>(x, lnw, lnb, xn, deg, accum);
    deg_kernel<<<(NEDGE + 255) / 256, 256, 0, stream>>>(ei, ew, deg);
    dinv_kernel<<<(N_NODES + 255) / 256, 256, 0, stream>>>(deg);
    gemm_kernel<<<N_NODES / 16, 256, 0, stream>>>(xn, Wg, Wl, h, out);
    scatter_kernel<<<(int)(((long long)NEDGE * 32) / 256), 256, 0, stream>>>(ei, ew, deg, h, accum);
    final_kernel<<<(N_NODES * D + 255) / 256, 256, 0, stream>>>(accum, h, deg, bg, bl, out);
}